// FasterRCNNLoss_893353197759
// MI455X (gfx1250) — compile-verified
//
#include <hip/hip_runtime.h>
#include <hip/hip_bf16.h>

typedef __attribute__((ext_vector_type(2))) float v2f;
typedef __attribute__((ext_vector_type(8))) float v8f;

#define B_   8
#define NBOX 16384
#define NGT  128
#define NCLS 91
#define ROWP 92   // padded row stride in LDS (91 classes + 1 pad slot)

// ---------------- init accumulators ----------------
__global__ void initAcc(float* __restrict__ acc) {
    acc[threadIdx.x] = 0.0f;   // 32 floats: acc[b*4 + {nvalid, npos, regsum, clssum}]
}

// ---------------- stage A: IoU match + labels + reg loss ----------------
__global__ __launch_bounds__(256) void stageA(const float* __restrict__ bbox,
                                              const int*   __restrict__ glabels,
                                              const float* __restrict__ gtbox,
                                              float* __restrict__ acc,
                                              int*   __restrict__ labels_out) {
    __shared__ float4 s_gt[NGT];
    __shared__ float  s_ga[NGT];
    __shared__ int    s_gl[NGT];
    __shared__ float  s_red[3];

    const int b    = blockIdx.x >> 6;   // 64 blocks / image
    const int tile = blockIdx.x & 63;
    const int t    = threadIdx.x;

    if (t < 3) s_red[t] = 0.0f;
    if (t < NGT) {
        float4 g = ((const float4*)gtbox)[b * NGT + t];
        s_gt[t] = g;
        s_ga[t] = fmaxf(g.z - g.x, 0.0f) * fmaxf(g.w - g.y, 0.0f);
        s_gl[t] = glabels[b * NGT + t];
    }
    __syncthreads();

    const int i = tile * 256 + t;                    // box index within image
    float4 bx = ((const float4*)bbox)[b * NBOX + i];
    const float a1 = fmaxf(bx.z - bx.x, 0.0f) * fmaxf(bx.w - bx.y, 0.0f);

    float binter = 0.0f, bunion = 1.0f;
    int   bidx   = 0;
    #pragma unroll 4
    for (int j = 0; j < NGT; ++j) {
        float4 g = s_gt[j];
        float lx = fmaxf(bx.x, g.x), ly = fmaxf(bx.y, g.y);
        float rx = fminf(bx.z, g.z), ry = fminf(bx.w, g.w);
        float w = fmaxf(rx - lx, 0.0f), h = fmaxf(ry - ly, 0.0f);
        float inter = w * h;
        float uni   = a1 + s_ga[j] - inter;
        // iou_j > iou_best  <=>  inter_j * U_best > inter_best * U_j  (U > 0)
        if (inter * fmaxf(bunion, 1e-9f) > binter * fmaxf(uni, 1e-9f)) {
            binter = inter; bunion = uni; bidx = j;
        }
    }
    const float miou = binter / fmaxf(bunion, 1e-9f);   // single division per box
    const bool pos = (miou >= 0.5f);
    const bool neg = (miou <  0.3f);
    const int  label = pos ? s_gl[bidx] : (neg ? 0 : -1);
    labels_out[b * NBOX + i] = label;

    float validf = (label != -1) ? 1.0f : 0.0f;
    float posf   = pos ? 1.0f : 0.0f;
    float reg    = 0.0f;
    if (pos) {
        float4 g = s_gt[bidx];
        float d, ad;
        d = bx.x - g.x; ad = fabsf(d); reg += (ad < 1.0f) ? 0.5f * d * d : ad - 0.5f;
        d = bx.y - g.y; ad = fabsf(d); reg += (ad < 1.0f) ? 0.5f * d * d : ad - 0.5f;
        d = bx.z - g.z; ad = fabsf(d); reg += (ad < 1.0f) ? 0.5f * d * d : ad - 0.5f;
        d = bx.w - g.w; ad = fabsf(d); reg += (ad < 1.0f) ? 0.5f * d * d : ad - 0.5f;
    }
    // wave32 reduce, then one LDS atomic per wave, one global atomic per block
    #pragma unroll
    for (int off = 16; off; off >>= 1) {
        validf += __shfl_down(validf, off, 32);
        posf   += __shfl_down(posf,   off, 32);
        reg    += __shfl_down(reg,    off, 32);
    }
    if ((t & 31) == 0) {
        atomicAdd(&s_red[0], validf);
        atomicAdd(&s_red[1], posf);
        atomicAdd(&s_red[2], reg);
    }
    __syncthreads();
    if (t == 0) {
        atomicAdd(&acc[b * 4 + 0], s_red[0]);
        atomicAdd(&acc[b * 4 + 1], s_red[1]);
        atomicAdd(&acc[b * 4 + 2], s_red[2]);
    }
}

// ---------------- stage B: logsumexp NLL via WMMA row-sums ----------------
// Block = 256 threads = 8 waves; each wave handles 16 boxes (block = 128 boxes).
__global__ __launch_bounds__(256) void stageB(const float* __restrict__ cls,
                                              const int*   __restrict__ labels,
                                              float* __restrict__ acc) {
    __shared__ float s_tile[8 * 16 * ROWP];   // 47104 B, stride-92 rows
    __shared__ float s_max[128];
    __shared__ float s_nll;

    const int b    = blockIdx.x >> 7;   // 128 blocks / image
    const int tile = blockIdx.x & 127;
    const int t    = threadIdx.x;
    const int w    = t >> 5;
    const int lane = t & 31;

    if (t == 0) s_nll = 0.0f;

    const int boxbase = tile * 128 + w * 16;     // first box of this wave (in image)
    const float* __restrict__ src = cls + ((size_t)b * NBOX + boxbase) * NCLS;
    float* __restrict__ row0 = s_tile + w * 16 * ROWP;

    // coalesced global read of 16*91 floats -> stride-92 LDS rows
    for (int x = lane; x < 16 * NCLS; x += 32) {
        int bxi = x / NCLS;
        int rem = x - bxi * NCLS;
        row0[bxi * ROWP + rem] = src[x];
    }
    if (lane < 16) row0[lane * ROWP + NCLS] = -1e30f;   // pad class 91 -> exp = 0
    __syncthreads();

    const int box  = lane & 15;
    const int half = lane >> 4;
    const float* __restrict__ row = row0 + box * ROWP;

    // per-box max (two lanes per box, halves of the class range)
    float m = -1e30f;
    const int c0 = half * 46;
    const int c1 = half ? NCLS : 46;
    for (int c = c0; c < c1; ++c) m = fmaxf(m, row[c]);
    m = fmaxf(m, __shfl_xor(m, 16, 32));
    s_max[w * 16 + box] = m;

    // sum_k exp(x-max): 23 accumulating V_WMMA_F32_16X16X4_F32 with all-ones B.
    // A layout (ISA 7.12.2): lanes 0-15 -> M=lane, K=0,1 ; lanes 16-31 -> M=lane-16, K=2,3
    v8f d = {};
    v2f ones = {1.0f, 1.0f};
    #pragma unroll
    for (int kb = 0; kb < ROWP; kb += 4) {
        v2f a;
        a.x = __expf(row[kb + 2 * half]     - m);
        a.y = __expf(row[kb + 2 * half + 1] - m);
        d = __builtin_amdgcn_wmma_f32_16x16x4_f32(false, a, false, ones,
                                                  (short)0, d, false, false);
    }

    // D layout: VGPR r holds M=r (lanes 0-15) and M=r+8 (lanes 16-31).
    // Lane 0 finalizes boxes 0..7, lane 16 finalizes boxes 8..15.
    if ((lane & 15) == 0) {
        float local = 0.0f;
        const int gbase = b * NBOX + boxbase + half * 8;
        #pragma unroll
        for (int r = 0; r < 8; ++r) {
            const int mb  = half * 8 + r;
            const int lab = labels[gbase + r];
            if (lab >= 0) {
                const float se = d[r];                      // >= 1, log is safe
                local += __logf(se) + s_max[w * 16 + mb] - row0[mb * ROWP + lab];
            }
        }
        atomicAdd(&s_nll, local);
    }
    __syncthreads();
    if (t == 0) atomicAdd(&acc[b * 4 + 3], s_nll);
}

// ---------------- finalize: per-image divides + batch sum ----------------
__global__ void finalize(const float* __restrict__ acc, float* __restrict__ out) {
    const int lane = threadIdx.x;
    float cls_l = 0.0f, reg_l = 0.0f;
    if (lane < B_) {
        const float nv = acc[lane * 4 + 0];
        const float np = acc[lane * 4 + 1];
        const float rs = acc[lane * 4 + 2];
        const float cs = acc[lane * 4 + 3];
        cls_l = cs / fmaxf(nv, 1.0f);
        reg_l = rs / fmaxf(np * 4.0f, 1.0f);
    }
    #pragma unroll
    for (int off = 4; off; off >>= 1) {
        cls_l += __shfl_down(cls_l, off, 32);
        reg_l += __shfl_down(reg_l, off, 32);
    }
    if (lane == 0) {
        out[0] = cls_l + reg_l;   // total
        out[1] = reg_l;           // regression_loss
        out[2] = cls_l;           // classification_loss
    }
}

extern "C" void kernel_launch(void* const* d_in, const int* in_sizes, int n_in,
                              void* d_out, int out_size, void* d_ws, size_t ws_size,
                              hipStream_t stream) {
    const float* cls  = (const float*)d_in[0];   // (8,16384,91) f32
    const float* bbox = (const float*)d_in[1];   // (8,16384,4)  f32
    const int*   glab = (const int*)  d_in[2];   // (8,128)      i32
    const float* gtb  = (const float*)d_in[3];   // (8,128,4)    f32
    float* out = (float*)d_out;                  // [total, reg, cls]

    float* acc    = (float*)d_ws;                       // 32 floats
    int*   labels = (int*)((char*)d_ws + 256);          // 8*16384 ints

    initAcc<<<1, 32, 0, stream>>>(acc);
    stageA<<<B_ * (NBOX / 256), 256, 0, stream>>>(bbox, glab, gtb, acc, labels);
    stageB<<<B_ * (NBOX / 128), 256, 0, stream>>>(cls, labels, acc);
    finalize<<<1, 32, 0, stream>>>(acc, out);
}